// ExampleTiedDropout_6786048327866
// MI455X (gfx1250) — compile-verified
//
#include <hip/hip_runtime.h>
#include <hip/hip_bf16.h>

// Problem constants from the reference: B=4096, C=32, H=W=8.
// Per-example element count = C*H*W = 2048; per-channel = H*W = 64.
#define ELEMS_PER_EX   2048
#define ELEMS_PER_CH   64
#define NCH            32
#define THREADS        256

typedef float v4f __attribute__((ext_vector_type(4)));
typedef __attribute__((address_space(1))) int GInt;
typedef __attribute__((address_space(3))) int LInt;

__global__ __launch_bounds__(THREADS)
void tied_dropout_gather_mul(const float* __restrict__ X,
                             const long long* __restrict__ idx,
                             const float* __restrict__ mask,
                             float* __restrict__ out)
{
    __shared__ float smask[NCH];

    const int b = blockIdx.x;
    const int t = threadIdx.x;

    // Uniform per-block gather index (scalarized by the compiler -> s_load).
    const long long id = idx[b];
    const float* mrow = mask + (size_t)id * ELEMS_PER_EX;

    // Stage the 32 per-channel mask scalars (mask is constant over H*W) into
    // LDS. Wave 0 (lanes 0..31, full EXEC) issues one async global->LDS load
    // per lane on the CDNA5 ASYNCcnt path, waits, then everyone barriers.
    if (t < NCH) {
#if __has_builtin(__builtin_amdgcn_global_load_async_to_lds_b32)
        __builtin_amdgcn_global_load_async_to_lds_b32(
            (GInt*)(mrow + t * ELEMS_PER_CH),
            (LInt*)&smask[t],
            /*imm offset=*/0, /*cpol=*/0);
#if __has_builtin(__builtin_amdgcn_s_wait_asynccnt)
        __builtin_amdgcn_s_wait_asynccnt(0);
#else
        asm volatile("s_wait_asynccnt 0" ::: "memory");
#endif
#else
        // Fallback: plain load + LDS store.
        smask[t] = mrow[t * ELEMS_PER_CH];
#endif
    }
    __syncthreads();

    // Each thread handles two float4 quads: one in the first half of the
    // example, one in the second half -> fully coalesced b128 across lanes.
    const size_t base = (size_t)b * ELEMS_PER_EX;
    const int e0 = t * 4;                 // [0, 1024)
    const int e1 = 1024 + t * 4;          // [1024, 2048)

    // Each quad lies inside one channel (64 elems/channel, e % 64 <= 60).
    const float m0 = smask[t >> 4];        // channel = e0 / 64
    const float m1 = smask[16 + (t >> 4)]; // channel = e1 / 64

    v4f r0, r1;

    if (m0 != 0.0f) {
        v4f x0 = __builtin_nontemporal_load((const v4f*)(X + base + e0));
        r0 = x0 * m0;
    } else {
        r0 = (v4f)(0.0f);                  // skip the X load entirely
    }

    if (m1 != 0.0f) {
        v4f x1 = __builtin_nontemporal_load((const v4f*)(X + base + e1));
        r1 = x1 * m1;
    } else {
        r1 = (v4f)(0.0f);
    }

    __builtin_nontemporal_store(r0, (v4f*)(out + base + e0));
    __builtin_nontemporal_store(r1, (v4f*)(out + base + e1));
}

extern "C" void kernel_launch(void* const* d_in, const int* in_sizes, int n_in,
                              void* d_out, int out_size, void* d_ws, size_t ws_size,
                              hipStream_t stream) {
    const float*     X    = (const float*)d_in[0];       // [B, 32, 8, 8] f32
    const long long* idxp = (const long long*)d_in[1];   // [B] int64
    const float*     mask = (const float*)d_in[2];       // [60000, 32, 8, 8] f32
    float*           out  = (float*)d_out;               // [B, 32, 8, 8] f32

    const int B = in_sizes[1];   // 4096 examples, one block per example
    (void)n_in; (void)out_size; (void)d_ws; (void)ws_size;

    tied_dropout_gather_mul<<<B, THREADS, 0, stream>>>(X, idxp, mask, out);
}